// NoisyGate_V2_40132174414261
// MI455X (gfx1250) — compile-verified
//
#include <hip/hip_runtime.h>
#include <math.h>

typedef __attribute__((ext_vector_type(2))) float v2f;
typedef __attribute__((ext_vector_type(4))) float v4f;
typedef __attribute__((ext_vector_type(8))) float v8f;

#define DN 768
#define EN 16
#define ROWS_PER_WAVE 16
#define WAVES_PER_BLOCK 8
#define ROWS_PER_BLOCK (ROWS_PER_WAVE * WAVES_PER_BLOCK)   // 128
#define TILE_PITCH 17                                      // avoid LDS bank conflicts

// ---------------------------------------------------------------------------
// Fused gating kernel: one wave owns a 16-row tile of inp.
//   D(16x16) += A(16x4, fp32) x B(4x16, fp32)   via V_WMMA_F32_16X16X4_F32
// A layout (ISA 7.12.2, 32-bit A 16x4): lane l -> M = l%16, K-base = 2*(l/16),
// so each lane loads a contiguous float2 from its row of inp.
// B layout: VGPR0 = rows K=0 (lanes 0-15) / K=2 (lanes 16-31); VGPR1 = K=1/K=3.
// We stage w_gate in LDS as float2 pairs s_wp[p*16+e] = {w[2p][e], w[2p+1][e]}
// so the B fragment is a single conflict-free ds_load_b64.
// ---------------------------------------------------------------------------
__global__ __launch_bounds__(256) void gate_kernel(const float* __restrict__ inp,
                                                   const float* __restrict__ wg,
                                                   float* __restrict__ gates,
                                                   float* __restrict__ g_imp,
                                                   float* __restrict__ g_load) {
  __shared__ v2f   s_wp[(DN / 2) * EN];                              // 48 KB
  __shared__ float s_tile[WAVES_PER_BLOCK][ROWS_PER_WAVE * TILE_PITCH]; // 8.5 KB
  __shared__ float s_imp[EN];
  __shared__ float s_load[EN];

  const int tid = threadIdx.x;
  if (tid < EN) { s_imp[tid] = 0.f; s_load[tid] = 0.f; }

  // Stage w_gate (768x16) into LDS as (K, K+1) float2 pairs. One-time 48 KB.
  for (int i = tid; i < (DN / 2) * EN; i += 256) {
    const int dp = i >> 4;
    const int e  = i & 15;
    v2f v;
    v.x = wg[(2 * dp) * EN + e];
    v.y = wg[(2 * dp + 1) * EN + e];
    s_wp[i] = v;
  }
  __syncthreads();

  const int wave = tid >> 5;
  const int lane = tid & 31;
  const int col  = lane & 15;   // M (for A) / N (for B) within the tile
  const int half = lane >> 4;   // selects K pair {0,1} vs {2,3}

  const size_t rowBase = (size_t)blockIdx.x * ROWS_PER_BLOCK + (size_t)wave * ROWS_PER_WAVE;
  const float* __restrict__ arow = inp + (rowBase + (size_t)col) * DN + 2 * half;

  // ---- streaming GEMM: 192 x V_WMMA_F32_16X16X4_F32, accumulate in v8f ----
  v8f acc = {};
#pragma unroll 8
  for (int kk = 0; kk < DN / 4; ++kk) {
    v2f a = *(const v2f*)(arow + 4 * kk);             // global_load_b64, contiguous per row
    v2f b = s_wp[(2 * kk + half) * EN + col];         // ds_load_b64, conflict-free
    acc = __builtin_amdgcn_wmma_f32_16x16x4_f32(
        /*neg_a=*/false, a, /*neg_b=*/false, b,
        /*c_mod=*/(short)0, acc, /*reuse_a=*/false, /*reuse_b=*/false);
  }

  // ---- spill 16x16 logit tile to LDS (C/D layout: VGPR j -> M=j / M=j+8) ----
  float* tile = &s_tile[wave][0];
#pragma unroll
  for (int j = 0; j < 8; ++j)
    tile[(j + 8 * half) * TILE_PITCH + col] = acc[j];
  __syncthreads();

  // ---- per-row softmax + top-2 (lanes 0..15 each own one row) ----
  float probs[EN];
  float t1 = -1.f, t2 = -1.f;
  int   i1 = 0,    i2 = 0;
  if (lane < 16) {
    const int r = lane;
    float mx = -3.4e38f;
#pragma unroll
    for (int e = 0; e < EN; ++e) {
      probs[e] = tile[r * TILE_PITCH + e];
      mx = fmaxf(mx, probs[e]);
    }
    float sum = 0.f;
#pragma unroll
    for (int e = 0; e < EN; ++e) { probs[e] = __expf(probs[e] - mx); sum += probs[e]; }
    const float inv = 1.f / sum;
#pragma unroll
    for (int e = 0; e < EN; ++e) probs[e] *= inv;
    // top-2 (first occurrence on ties, matching top_k ordering)
#pragma unroll
    for (int e = 0; e < EN; ++e) if (probs[e] > t1) { t1 = probs[e]; i1 = e; }
#pragma unroll
    for (int e = 0; e < EN; ++e) if (e != i1 && probs[e] > t2) { t2 = probs[e]; i2 = e; }
  }
  __syncthreads();   // logits fully consumed before tile is overwritten

  if (lane < 16) {
    const int r = lane;
    // sparse gates row -> global (4x b128 stores)
    float g[EN];
#pragma unroll
    for (int e = 0; e < EN; ++e) g[e] = (e == i1) ? t1 : ((e == i2) ? t2 : 0.f);
    float* orow = gates + (rowBase + (size_t)r) * EN;
#pragma unroll
    for (int e = 0; e < EN; e += 4) {
      v4f o = { g[e], g[e + 1], g[e + 2], g[e + 3] };
      *(v4f*)(orow + e) = o;
    }
    // probs back into tile for the per-expert column reduction
#pragma unroll
    for (int e = 0; e < EN; ++e) tile[r * TILE_PITCH + e] = probs[e];
    // load counts: 2 LDS float atomics per row
    atomicAdd(&s_load[i1], 1.f);
    atomicAdd(&s_load[i2], 1.f);
  }
  __syncthreads();

  // ---- importance: lane e sums its wave's probs column e ----
  if (lane < 16) {
    float s = 0.f;
#pragma unroll
    for (int r = 0; r < ROWS_PER_WAVE; ++r) s += tile[r * TILE_PITCH + lane];
    atomicAdd(&s_imp[lane], s);
  }
  __syncthreads();

  // ---- 32 global atomics per block (1024 blocks -> 32K total) ----
  if (tid < EN) {
    atomicAdd(&g_imp[tid], s_imp[tid]);
    atomicAdd(&g_load[tid], s_load[tid]);
  }
}

// Zero the workspace accumulators (ws is poisoned; graph replays must be
// deterministic, so re-zero every launch).
__global__ void zero_ws_kernel(float* ws) {
  if (threadIdx.x < 2 * EN) ws[threadIdx.x] = 0.f;
}

__device__ inline float cv_sq(const float* x) {
  float s = 0.f;
#pragma unroll
  for (int e = 0; e < EN; ++e) s += x[e];
  const float mean = s * (1.f / EN);
  float v = 0.f;
#pragma unroll
  for (int e = 0; e < EN; ++e) { const float d = x[e] - mean; v += d * d; }
  v *= (1.f / (EN - 1));           // ddof=1
  return v / (mean * mean + 1e-10f);
}

__global__ void loss_kernel(const float* __restrict__ g_imp,
                            const float* __restrict__ g_load,
                            float* __restrict__ loss_out) {
  if (threadIdx.x == 0) {
    float imp[EN], ld[EN];
#pragma unroll
    for (int e = 0; e < EN; ++e) { imp[e] = g_imp[e]; ld[e] = g_load[e]; }
    imp[0] *= 6.f;
    imp[1] *= 4.f;
    *loss_out = cv_sq(imp) + cv_sq(ld);
  }
}

extern "C" void kernel_launch(void* const* d_in, const int* in_sizes, int n_in,
                              void* d_out, int out_size, void* d_ws, size_t ws_size,
                              hipStream_t stream) {
  const float* inp = (const float*)d_in[0];   // (N, 768) fp32
  const float* wg  = (const float*)d_in[1];   // (768, 16) fp32
  const int N = in_sizes[0] / DN;             // 131072

  float* gates  = (float*)d_out;              // N*16 gate values
  float* loss   = gates + (size_t)N * EN;     // + scalar loss
  float* g_imp  = (float*)d_ws;               // [16] importance accum
  float* g_load = g_imp + EN;                 // [16] load accum

  zero_ws_kernel<<<1, 32, 0, stream>>>(g_imp);
  gate_kernel<<<N / ROWS_PER_BLOCK, 256, 0, stream>>>(inp, wg, gates, g_imp, g_load);
  loss_kernel<<<1, 32, 0, stream>>>(g_imp, g_load, loss);
}